// UnifiedResidueGeometry_72327249265114
// MI455X (gfx1250) — compile-verified
//
#include <hip/hip_runtime.h>
#include <math.h>

typedef __attribute__((ext_vector_type(2))) float v2f;
typedef __attribute__((ext_vector_type(8))) float v8f;

#define HID    96
#define HEADS  4
#define DH     24
#define NRES   2048
#define BATCH  2
#define FEAT   124          // HID + 28 spatial
#define NEG_BIG (-1e30f)

static __device__ __forceinline__ v2f ld2(const float* p) { return *(const v2f*)p; }

// D = A(16x4 f32) * B(4x16 f32) + C(16x16 f32), CDNA5 wave32 WMMA.
// A frag per lane (r=lane&15, c0=2*(lane>>4)): {A[r][k0+c0], A[r][k0+c0+1]}
// B frag per lane:                              {B[k0+c0][r], B[k0+c0+1][r]}
// C/D per lane: 8 VGPRs, element v -> C[v + 8*(lane>>4)][lane&15]
static __device__ __forceinline__ v8f wmma_f32(v2f a, v2f b, v8f c) {
  return __builtin_amdgcn_wmma_f32_16x16x4_f32(false, a, false, b, (short)0, c,
                                               false, false);
}

// ---------------------------------------------------------------------------
// Kernel 1: Q/K/V projections  [4096,96] @ [96,96] + bias, via f32 WMMA.
// One wave per 16-row tile; blockIdx.y selects which of the 3 weight sets.
// ---------------------------------------------------------------------------
__global__ void __launch_bounds__(128)
qkv_gemm_kernel(const float* __restrict__ X,
                const float* __restrict__ Wq, const float* __restrict__ bq,
                const float* __restrict__ Wk, const float* __restrict__ bk,
                const float* __restrict__ Wv, const float* __restrict__ bv,
                float* __restrict__ qbuf, float* __restrict__ kbuf,
                float* __restrict__ vbuf) {
  const int lane = threadIdx.x & 31;
  const int w    = threadIdx.x >> 5;
  const int r    = lane & 15;
  const int g    = lane >> 4;
  const int c0   = 2 * g;
  const int tile = blockIdx.x * 4 + w;       // 0..255
  const int rowbase = tile * 16;

  const float *W, *bias; float* dst;
  if (blockIdx.y == 0)      { W = Wq; bias = bq; dst = qbuf; }
  else if (blockIdx.y == 1) { W = Wk; bias = bk; dst = kbuf; }
  else                      { W = Wv; bias = bv; dst = vbuf; }

  // Preload the A fragments for the whole 16x96 input tile (24 K-chunks of 4)
  v2f af[24];
  const float* xrow = X + (size_t)(rowbase + r) * HID + c0;
#pragma unroll
  for (int j = 0; j < 24; ++j) af[j] = ld2(xrow + 4 * j);

  for (int ct = 0; ct < 6; ++ct) {           // 6 column tiles of 16
    const int col = ct * 16 + r;
    v8f acc = {};
#pragma unroll
    for (int j = 0; j < 24; ++j) {
      v2f bf;
      bf.x = W[(size_t)(4 * j + c0) * HID + col];
      bf.y = W[(size_t)(4 * j + c0 + 1) * HID + col];
      acc = wmma_f32(af[j], bf, acc);
    }
    const float bcol = bias[col];
#pragma unroll
    for (int v = 0; v < 8; ++v) {
      const int m = v + 8 * g;
      dst[(size_t)(rowbase + m) * HID + col] = acc[v] + bcol;
    }
  }
}

// ---------------------------------------------------------------------------
// Kernel 2: fused flash attention + geometric epilogue.
// One wave per (batch, head, 16-query tile). Online softmax over 128 key
// tiles; P@Vext (V | pos_CA) accumulated with WMMA; alpha.CA gives bias
// (since softmax rows sum to 1: bias = pos_CB - alpha@pos_CA).
// ---------------------------------------------------------------------------
__global__ void __launch_bounds__(128)
attention_kernel(const float* __restrict__ qbuf,
                 const float* __restrict__ kbuf,
                 const float* __restrict__ vbuf,
                 const float* __restrict__ posCA,
                 const float* __restrict__ posCB,
                 const unsigned char* __restrict__ mask,
                 float* __restrict__ feat) {
  __shared__ float pbuf[4][16][18];          // per-wave 16x16 P staging (+pad)
  const int lane = threadIdx.x & 31;
  const int w    = threadIdx.x >> 5;
  const int r    = lane & 15;
  const int g    = lane >> 4;
  const int c0   = 2 * g;
  const int widx = blockIdx.x * 4 + w;       // 0..1023
  const int b    = widx >> 9;
  const int head = (widx >> 7) & 3;
  const int qt   = widx & 127;
  const int qbase = qt * 16;
  const size_t rowoff = (size_t)b * NRES;
  const int hoff = head * DH;

  // Q fragments for this tile (6 K-chunks over dh=24)
  v2f qf[6];
  {
    const float* qrow = qbuf + (rowoff + qbase + r) * HID + hoff + c0;
#pragma unroll
    for (int j = 0; j < 6; ++j) qf[j] = ld2(qrow + 4 * j);
  }

  float mrow[8], lrow[8];
#pragma unroll
  for (int v = 0; v < 8; ++v) { mrow[v] = NEG_BIG; lrow[v] = 0.f; }
  v8f accA = {};                             // Vext cols 0..15
  v8f accB = {};                             // Vext cols 16..26 (27..31 = 0)

  // Per-lane column source for the high output tile: colB = 16 + r
  const int colB  = 16 + r;
  const int bmode = (colB < DH) ? 0 : (colB < DH + 3 ? 1 : 2);
  const int bcomp = colB - DH;               // pos_CA component when bmode==1

  for (int kt = 0; kt < NRES / 16; ++kt) {
    const int kbase = kt * 16;
    if (kt + 1 < NRES / 16)                  // prefetch next K tile rows
      __builtin_prefetch(kbuf + (rowoff + kbase + 16 + r) * HID + hoff + c0, 0, 1);

    // S = Q K^T : rows = queries (C/D rows), cols = key index in tile
    v8f s = {};
    const float* krow = kbuf + (rowoff + kbase + r) * HID + hoff + c0;
#pragma unroll
    for (int j = 0; j < 6; ++j) s = wmma_f32(qf[j], ld2(krow + 4 * j), s);

    // key mask (this lane owns key column kbase + r)
    if (!mask[rowoff + kbase + r]) {
#pragma unroll
      for (int v = 0; v < 8; ++v) s[v] = NEG_BIG;
    }

    // Online softmax: row stats reduced across the 16-lane half-wave.
#pragma unroll
    for (int v = 0; v < 8; ++v) {
      float x  = s[v];
      float xm = x;
      xm = fmaxf(xm, __shfl_xor(xm, 1, 32));
      xm = fmaxf(xm, __shfl_xor(xm, 2, 32));
      xm = fmaxf(xm, __shfl_xor(xm, 4, 32));
      xm = fmaxf(xm, __shfl_xor(xm, 8, 32));
      const float mnew  = fmaxf(mrow[v], xm);
      const float scale = __expf(mrow[v] - mnew);
      const float p     = __expf(x - mnew);
      float ps = p;
      ps += __shfl_xor(ps, 1, 32);
      ps += __shfl_xor(ps, 2, 32);
      ps += __shfl_xor(ps, 4, 32);
      ps += __shfl_xor(ps, 8, 32);
      lrow[v] = lrow[v] * scale + ps;
      mrow[v] = mnew;
      s[v]    = p;
      accA[v] *= scale;
      accB[v] *= scale;
    }

    // Stage P (C/D layout) to LDS, reload in A layout. Wave-local: only
    // need the CDNA5 DS counter to drain, no block barrier.
#pragma unroll
    for (int v = 0; v < 8; ++v) pbuf[w][v + 8 * g][r] = s[v];
    asm volatile("s_wait_dscnt 0x0" ::: "memory");

#pragma unroll
    for (int t = 0; t < 4; ++t) {
      v2f pa;
      pa.x = pbuf[w][r][4 * t + c0];
      pa.y = pbuf[w][r][4 * t + c0 + 1];
      const size_t row0 = rowoff + kbase + 4 * t + c0;
      v2f ba;                                // Vext cols 0..15 (always V)
      ba.x = vbuf[row0 * HID + hoff + r];
      ba.y = vbuf[(row0 + 1) * HID + hoff + r];
      v2f bb;                                // Vext cols 16..31
      if (bmode == 0) {
        bb.x = vbuf[row0 * HID + hoff + colB];
        bb.y = vbuf[(row0 + 1) * HID + hoff + colB];
      } else if (bmode == 1) {
        bb.x = posCA[row0 * 3 + bcomp];
        bb.y = posCA[(row0 + 1) * 3 + bcomp];
      } else { bb.x = 0.f; bb.y = 0.f; }
      accA = wmma_f32(pa, ba, accA);
      accB = wmma_f32(pa, bb, accB);
    }
    asm volatile("" ::: "memory");
  }

  // Normalize by the softmax denominator.
#pragma unroll
  for (int v = 0; v < 8; ++v) {
    const float inv = 1.0f / lrow[v];
    accA[v] *= inv;
    accB[v] *= inv;
  }

  // feat_node: this head's 24 columns inside feat_all[0..95].
  float* fb = feat + (rowoff + qbase) * FEAT;
#pragma unroll
  for (int v = 0; v < 8; ++v) {
    const int m = v + 8 * g;
    fb[(size_t)m * FEAT + hoff + r] = accA[v];
    if (colB < DH) fb[(size_t)m * FEAT + hoff + colB] = accB[v];
  }

  // Stage normalized alpha@pos_CA (cols 24..26 live at r=8,9,10) via LDS.
#pragma unroll
  for (int v = 0; v < 8; ++v) pbuf[w][v + 8 * g][r] = accB[v];
  asm volatile("s_wait_dscnt 0x0" ::: "memory");

  if (lane < 16) {                           // one lane per query row
    const int m = lane;
    const size_t res = rowoff + qbase + m;
    const int nidx = qbase + m;
    const float aCAx = pbuf[w][m][8], aCAy = pbuf[w][m][9], aCAz = pbuf[w][m][10];
    const float cbx = posCB[res * 3 + 0], cby = posCB[res * 3 + 1], cbz = posCB[res * 3 + 2];
    const float cax = posCA[res * 3 + 0], cay = posCA[res * 3 + 1], caz = posCA[res * 3 + 2];
    const float bx = cbx - aCAx, by = cby - aCAy, bz = cbz - aCAz;
    const float dist = sqrtf(bx * bx + by * by + bz * bz);

    // Build per-residue frame (columns e1n,e2n,e3), identity if degenerate/last.
    float e1x = cbx - cax, e1y = cby - cay, e1z = cbz - caz;
    const float n1 = sqrtf(e1x * e1x + e1y * e1y + e1z * e1z);
    const bool valid1 = n1 > 1e-6f;
    const float i1 = 1.0f / fmaxf(n1, 1e-12f);
    e1x *= i1; e1y *= i1; e1z *= i1;
    float e2x = e1y, e2y = -e1x, e2z = 0.f;          // cross(e1n, z)
    if (sqrtf(e2x * e2x + e2y * e2y) < 1e-6f) {      // cross(e1n, y)
      e2x = -e1z; e2y = 0.f; e2z = e1x;
    }
    const float n2 = sqrtf(e2x * e2x + e2y * e2y + e2z * e2z);
    const bool valid2 = n2 > 1e-6f;
    const float i2 = 1.0f / fmaxf(n2, 1e-12f);
    e2x *= i2; e2y *= i2; e2z *= i2;
    const float e3x = e1y * e2z - e1z * e2y;
    const float e3y = e1z * e2x - e1x * e2z;
    const float e3z = e1x * e2y - e1y * e2x;
    const bool valid = valid1 && valid2 && (nidx < NRES - 1);

    float px, py, pz;
    if (valid) {
      px = e1x * bx + e2x * by + e3x * bz;
      py = e1y * bx + e2y * by + e3y * bz;
      pz = e1z * bx + e2z * by + e3z * bz;
    } else { px = bx; py = by; pz = bz; }
    const float pn = sqrtf(px * px + py * py + pz * pz) + 1e-10f;

    float* frow = feat + res * FEAT;
    frow[HID + head * 3 + 0] = px;
    frow[HID + head * 3 + 1] = py;
    frow[HID + head * 3 + 2] = pz;
    frow[HID + 12 + head]    = dist;
    frow[HID + 16 + head * 3 + 0] = px / pn;
    frow[HID + 16 + head * 3 + 1] = py / pn;
    frow[HID + 16 + head * 3 + 2] = pz / pn;
  }
}

// ---------------------------------------------------------------------------
// Kernel 3: out = LN2(x + mask*relu(LN1(feat_all @ Wo + bo)))
// One wave per row; each lane owns 3 of the 96 output channels; layernorm
// stats via wave32 butterfly shuffles.
// ---------------------------------------------------------------------------
__global__ void __launch_bounds__(256)
output_kernel(const float* __restrict__ feat,
              const float* __restrict__ X,
              const unsigned char* __restrict__ mask,
              const float* __restrict__ Wo, const float* __restrict__ bo,
              const float* __restrict__ g1, const float* __restrict__ b1,
              const float* __restrict__ g2, const float* __restrict__ b2,
              float* __restrict__ out) {
  __shared__ float frow[8][128];
  const int lane = threadIdx.x & 31;
  const int w    = threadIdx.x >> 5;
  const size_t row = (size_t)blockIdx.x * 8 + w;   // 0..4095

#pragma unroll
  for (int t = 0; t < 4; ++t) {
    const int i = lane + 32 * t;
    frow[w][i] = (i < FEAT) ? feat[row * FEAT + i] : 0.f;
  }
  asm volatile("s_wait_dscnt 0x0" ::: "memory");

  float y0 = bo[lane], y1 = bo[lane + 32], y2 = bo[lane + 64];
  for (int i = 0; i < FEAT; ++i) {
    const float f = frow[w][i];
    const float* wr = Wo + (size_t)i * HID;
    y0 = fmaf(f, wr[lane], y0);
    y1 = fmaf(f, wr[lane + 32], y1);
    y2 = fmaf(f, wr[lane + 64], y2);
  }
  float s = y0 + y1 + y2, q = y0 * y0 + y1 * y1 + y2 * y2;
#pragma unroll
  for (int off = 16; off >= 1; off >>= 1) {
    s += __shfl_xor(s, off, 32);
    q += __shfl_xor(q, off, 32);
  }
  const float mean = s * (1.0f / 96.0f);
  const float var  = q * (1.0f / 96.0f) - mean * mean;
  const float inv  = rsqrtf(var + 1e-5f);
  float h0 = fmaxf((y0 - mean) * inv * g1[lane]      + b1[lane],      0.f);
  float h1 = fmaxf((y1 - mean) * inv * g1[lane + 32] + b1[lane + 32], 0.f);
  float h2 = fmaxf((y2 - mean) * inv * g1[lane + 64] + b1[lane + 64], 0.f);
  if (!mask[row]) { h0 = 0.f; h1 = 0.f; h2 = 0.f; }

  const float r0 = X[row * HID + lane]      + h0;
  const float r1 = X[row * HID + lane + 32] + h1;
  const float r2 = X[row * HID + lane + 64] + h2;
  float s2 = r0 + r1 + r2, q2 = r0 * r0 + r1 * r1 + r2 * r2;
#pragma unroll
  for (int off = 16; off >= 1; off >>= 1) {
    s2 += __shfl_xor(s2, off, 32);
    q2 += __shfl_xor(q2, off, 32);
  }
  const float mean2 = s2 * (1.0f / 96.0f);
  const float var2  = q2 * (1.0f / 96.0f) - mean2 * mean2;
  const float inv2  = rsqrtf(var2 + 1e-5f);
  out[row * HID + lane]      = (r0 - mean2) * inv2 * g2[lane]      + b2[lane];
  out[row * HID + lane + 32] = (r1 - mean2) * inv2 * g2[lane + 32] + b2[lane + 32];
  out[row * HID + lane + 64] = (r2 - mean2) * inv2 * g2[lane + 64] + b2[lane + 64];
}

// ---------------------------------------------------------------------------
extern "C" void kernel_launch(void* const* d_in, const int* in_sizes, int n_in,
                              void* d_out, int out_size, void* d_ws, size_t ws_size,
                              hipStream_t stream) {
  const float* X     = (const float*)d_in[0];
  const float* posCA = (const float*)d_in[1];
  const float* posCB = (const float*)d_in[2];
  const unsigned char* mask = (const unsigned char*)d_in[3];
  const float* Wq = (const float*)d_in[4];
  const float* bq = (const float*)d_in[5];
  const float* Wk = (const float*)d_in[6];
  const float* bk = (const float*)d_in[7];
  const float* Wv = (const float*)d_in[8];
  const float* bv = (const float*)d_in[9];
  const float* Wo = (const float*)d_in[10];
  const float* bo = (const float*)d_in[11];
  const float* g1 = (const float*)d_in[12];
  const float* b1 = (const float*)d_in[13];
  const float* g2 = (const float*)d_in[14];
  const float* b2 = (const float*)d_in[15];

  const size_t rows = (size_t)BATCH * NRES;        // 4096
  float* ws   = (float*)d_ws;
  float* qbuf = ws;
  float* kbuf = qbuf + rows * HID;
  float* vbuf = kbuf + rows * HID;
  float* feat = vbuf + rows * HID;                 // rows * FEAT floats

  qkv_gemm_kernel<<<dim3(64, 3), 128, 0, stream>>>(X, Wq, bq, Wk, bk, Wv, bv,
                                                   qbuf, kbuf, vbuf);
  attention_kernel<<<256, 128, 0, stream>>>(qbuf, kbuf, vbuf, posCA, posCB,
                                            mask, feat);
  output_kernel<<<512, 256, 0, stream>>>(feat, X, mask, Wo, bo, g1, b1, g2, b2,
                                         (float*)d_out);
}